// GraphSAGE_61065845015011
// MI455X (gfx1250) — compile-verified
//
#include <hip/hip_runtime.h>

typedef __attribute__((ext_vector_type(2))) float v2f;
typedef __attribute__((ext_vector_type(8))) float v8f;

#define CH 128   // IN_CH == HID_CH == 128 per reference

// ---------------------------------------------------------------------------
// Degree count: cnt[dst] += 1  (float so the same buffer becomes 1/deg later)
// ---------------------------------------------------------------------------
__global__ __launch_bounds__(256)
void deg_count_kernel(const int* __restrict__ dst, float* __restrict__ cnt, int E) {
    int e = blockIdx.x * blockDim.x + threadIdx.x;
    if (e < E) atomicAdd(&cnt[dst[e]], 1.0f);
}

__global__ __launch_bounds__(256)
void deg_inv_kernel(float* __restrict__ cnt, int M) {
    int i = blockIdx.x * blockDim.x + threadIdx.x;
    if (i < M) cnt[i] = 1.0f / fmaxf(cnt[i], 1.0f);
}

// ---------------------------------------------------------------------------
// Edge scatter: one wave per edge. 32 lanes read the 128-float source row as
// coalesced float4 (512B burst) and atomically add into sum[dst] (hot rows
// live in the 192MB L2, so global_atomic_add_f32 stays L2-local).
// ---------------------------------------------------------------------------
__global__ __launch_bounds__(256)
void scatter_sum_kernel(const float* __restrict__ feat,
                        const int* __restrict__ src,
                        const int* __restrict__ dst,
                        float* __restrict__ sum, int E) {
    const int lane = threadIdx.x & 31;
    const int wave = threadIdx.x >> 5;
    const int e = blockIdx.x * 8 + wave;
    if (e >= E) return;
    const int s = src[e];
    const int d = dst[e];
    const float4 v = ((const float4*)(feat + (size_t)s * CH))[lane];
    float* o = sum + (size_t)d * CH + lane * 4;
    atomicAdd(o + 0, v.x);
    atomicAdd(o + 1, v.y);
    atomicAdd(o + 2, v.z);
    atomicAdd(o + 3, v.w);
}

// ---------------------------------------------------------------------------
// Fused SAGE GEMM via V_WMMA_F32_16X16X4_F32 (exact fp32 matrix path):
//   out[m,n] = act( bias[n] + sum_k (A1[m,k]*rowscale[m])*W1[n,k]
//                            + sum_k  A2[m,k]*W2[n,k] )
// One wave = one 16x16 C tile. K stepped by 4.
// A 16x4 layout (ISA 7.12.2): lane = M | (K>=2)<<4 ; vgpr = K&1
//   -> lane (m = lane&15) loads float2 at column k0 + 2*(lane>>4).
// B 4x16 layout mirrors C/D striping: lane = N | (K>=2)<<4.
// C/D: vgpr r holds row r (lanes 0-15) / row r+8 (lanes 16-31), col = lane&15.
// Tile predicate is wave-uniform => EXEC all ones inside WMMA (required).
// ---------------------------------------------------------------------------
__global__ __launch_bounds__(256)
void sage_wmma_gemm_kernel(const float* __restrict__ A1,
                           const float* __restrict__ rowscale,  // nullable
                           const float* __restrict__ W1,        // [N,K] row-major
                           const float* __restrict__ A2,        // nullable
                           const float* __restrict__ W2,        // nullable
                           const float* __restrict__ bias,
                           float* __restrict__ out,
                           int M, int N, int K, int doRelu) {
    const int lane = threadIdx.x & 31;
    const int wave = threadIdx.x >> 5;
    const int colTiles = N >> 4;
    const int totalTiles = (M >> 4) * colTiles;
    const int tile = blockIdx.x * 8 + wave;      // wave-uniform
    if (tile >= totalTiles) return;              // whole-wave exit: EXEC stays full

    const int m0 = (tile / colTiles) << 4;
    const int n0 = (tile % colTiles) << 4;

    const int mrow  = m0 + (lane & 15);          // A row this lane feeds
    const int nrow  = n0 + (lane & 15);          // W row this lane feeds
    const int khalf = (lane >> 4) << 1;          // 0 for lanes 0-15, 2 for 16-31

    float scale = 1.0f;
    if (rowscale) scale = rowscale[mrow];        // fold mean-divide into A load

    v8f c = {};
    {
        const float* ap = A1 + (size_t)mrow * K;
        const float* wp = W1 + (size_t)nrow * K;
        for (int k0 = 0; k0 < K; k0 += 4) {
            v2f a, b;
            a.x = ap[k0 + khalf + 0] * scale;
            a.y = ap[k0 + khalf + 1] * scale;
            b.x = wp[k0 + khalf + 0];
            b.y = wp[k0 + khalf + 1];
            c = __builtin_amdgcn_wmma_f32_16x16x4_f32(
                    false, a, false, b, (short)0, c, false, false);
        }
    }
    if (A2) {
        const float* ap = A2 + (size_t)mrow * K;
        const float* wp = W2 + (size_t)nrow * K;
        for (int k0 = 0; k0 < K; k0 += 4) {
            v2f a, b;
            a.x = ap[k0 + khalf + 0];
            a.y = ap[k0 + khalf + 1];
            b.x = wp[k0 + khalf + 0];
            b.y = wp[k0 + khalf + 1];
            c = __builtin_amdgcn_wmma_f32_16x16x4_f32(
                    false, a, false, b, (short)0, c, false, false);
        }
    }

    const int ccol  = n0 + (lane & 15);
    const int rbase = m0 + ((lane >> 4) << 3);
    const float bv = bias ? bias[ccol] : 0.0f;
#pragma unroll
    for (int r = 0; r < 8; ++r) {
        float v = c[r] + bv;
        if (doRelu) v = fmaxf(v, 0.0f);
        out[(size_t)(rbase + r) * N + ccol] = v;
    }
}

// ---------------------------------------------------------------------------
// Host orchestration (graph-capture safe: only async ops on `stream`).
// Inputs: x, edge_index, Wl1, bl1, Wr1, Wl2, bl2, Wr2, Wlin, blin
// ---------------------------------------------------------------------------
extern "C" void kernel_launch(void* const* d_in, const int* in_sizes, int n_in,
                              void* d_out, int out_size, void* d_ws, size_t ws_size,
                              hipStream_t stream) {
    const float* x    = (const float*)d_in[0];
    const int*   ei   = (const int*)d_in[1];
    const float* Wl1  = (const float*)d_in[2];
    const float* bl1  = (const float*)d_in[3];
    const float* Wr1  = (const float*)d_in[4];
    const float* Wl2  = (const float*)d_in[5];
    const float* bl2  = (const float*)d_in[6];
    const float* Wr2  = (const float*)d_in[7];
    const float* Wlin = (const float*)d_in[8];
    const float* blin = (const float*)d_in[9];

    const int M    = in_sizes[0] / CH;   // 50000 (divisible by 16)
    const int E    = in_sizes[1] / 2;    // 800000
    const int OUTC = in_sizes[9];        // 64
    const int* src = ei;
    const int* dst = ei + E;

    const size_t nodeFeats = (size_t)M * CH;
    float* sum1 = (float*)d_ws;
    float* h1   = sum1 + nodeFeats;
    float* sum2 = h1   + nodeFeats;
    float* h2   = sum2 + nodeFeats;
    float* deg  = h2   + nodeFeats;      // degree -> 1/max(deg,1)

    const int edgeBlksScalar = (E + 255) / 256;
    const int edgeBlksWave   = (E + 7) / 8;          // 8 waves/block, 1 edge/wave
    const int gemmBlksHid    = ((M / 16) * (CH  / 16) + 7) / 8;
    const int gemmBlksOut    = ((M / 16) * (OUTC / 16) + 7) / 8;

    // ---- degree (shared by both layers) ----
    hipMemsetAsync(deg, 0, (size_t)M * sizeof(float), stream);
    deg_count_kernel<<<edgeBlksScalar, 256, 0, stream>>>(dst, deg, E);
    deg_inv_kernel<<<(M + 255) / 256, 256, 0, stream>>>(deg, M);

    // ---- layer 1: h1 = relu(mean(x)@Wl1^T + bl1 + x@Wr1^T) ----
    hipMemsetAsync(sum1, 0, nodeFeats * sizeof(float), stream);
    scatter_sum_kernel<<<edgeBlksWave, 256, 0, stream>>>(x, src, dst, sum1, E);
    sage_wmma_gemm_kernel<<<gemmBlksHid, 256, 0, stream>>>(
        sum1, deg, Wl1, x, Wr1, bl1, h1, M, CH, CH, 1);

    // ---- layer 2: h2 = relu(mean(h1)@Wl2^T + bl2 + h1@Wr2^T) ----
    hipMemsetAsync(sum2, 0, nodeFeats * sizeof(float), stream);
    scatter_sum_kernel<<<edgeBlksWave, 256, 0, stream>>>(h1, src, dst, sum2, E);
    sage_wmma_gemm_kernel<<<gemmBlksHid, 256, 0, stream>>>(
        sum2, deg, Wl2, h1, Wr2, bl2, h2, M, CH, CH, 1);

    // ---- output: out = h2@Wlin^T + blin ----
    sage_wmma_gemm_kernel<<<gemmBlksOut, 256, 0, stream>>>(
        h2, nullptr, Wlin, nullptr, nullptr, blin, (float*)d_out, M, OUTC, CH, 0);
}